// VarConvND_89816356094302
// MI455X (gfx1250) — compile-verified
//
#include <hip/hip_runtime.h>

typedef __attribute__((ext_vector_type(2)))  float        v2f;
typedef __attribute__((ext_vector_type(8)))  float        v8f;
typedef __attribute__((ext_vector_type(4)))  unsigned int v4u;
typedef __attribute__((ext_vector_type(4)))  int          v4i;
typedef __attribute__((ext_vector_type(8)))  int          v8i;

#define C_IN    128
#define C_OUT   64
#define BATCH   64
#define S_TOT   1024
#define K_TOT   1152            // 128 * 9
#define NCH     8               // channels per K-chunk
#define KC      72              // NCH * 9 k-values per chunk
#define KCP     74              // A LDS row stride (even -> aligned b64 frags)
#define WROW    72              // TDM-padded weight row stride in floats (64 + 8 pad)
#define NCHUNK  16              // K_TOT / KC

// Issue a TDM load of one 72(k) x 64(c) f32 weight chunk into LDS.
// D# per CDNA5 ISA:
//   group0: count=1 | lds_addr | global_addr | type=2
//   group1: data_size=4B, pad_enable, pad_interval=64 DW, pad_amount=8 DW,
//           tensor_dim0=64, tensor_dim1=72, tile_dim0=64, tile_dim1=72,
//           tensor_dim0_stride=64
//   groups 2/3 (and trailing group): zero (2-D tile, higher dims unused)
__device__ __forceinline__ void tdm_load_w_chunk(const float* gsrc, unsigned int lds_off) {
    const unsigned long long ga = (unsigned long long)(uintptr_t)gsrc;
    v4u g0;
    g0[0] = 1u;                                      // count=1 (valid user D#)
    g0[1] = lds_off;                                 // lds_addr (bytes)
    g0[2] = (unsigned int)ga;                        // global_addr[31:0]
    g0[3] = ((unsigned int)(ga >> 32) & 0x01FFFFFFu) // global_addr[56:32]
            | (2u << 30);                            // type=2 ("image")
    v8i g1;
    g1[0] = (int)((2u << 16)     // data_size = 4 bytes
                | (1u << 20)     // pad_enable
                | (5u << 22)     // pad_interval: 64 DWORDs (one row)
                | (7u << 25));   // pad_amount: 8 DWORDs -> row stride 72 floats
    g1[1] = (int)(64u << 16);    // tensor_dim0 = 64  (bits 79:48)
    g1[2] = (int)(72u << 16);    // tensor_dim1 = 72  (bits 111:80)
    g1[3] = (int)(64u << 16);    // tile_dim0  = 64  (bits 127:112)
    g1[4] = 72;                  // tile_dim1  = 72, tile_dim2 = 0
    g1[5] = 64;                  // tensor_dim0_stride = 64 (low 32)
    g1[6] = 0;                   // stride0 hi / stride1 lo
    g1[7] = 0;
    const v4i z4 = {0, 0, 0, 0};
    const v8i z8 = {0, 0, 0, 0, 0, 0, 0, 0};
    __builtin_amdgcn_tensor_load_to_lds(g0, g1, z4, z4, z8, 0);
}

__global__ __launch_bounds__(512, 1)
void varconv_wmma_f32(const float* __restrict__ X,
                      const float* __restrict__ W,
                      const float* __restrict__ Bias,
                      float* __restrict__ Out)
{
    __shared__ float Alds[BATCH][KCP];        // unfolded activations + bias, [m][k]
    __shared__ float Wbuf[2][KC * WROW];      // TDM double buffer, [k][c] stride 72

    const int s = blockIdx.x;                 // spatial location
    const int y = s >> 5;
    const int x = s & 31;
    const float bias_s = Bias[s];

    const int t    = threadIdx.x;             // 0..511, 16 waves
    const int wave = t >> 5;
    const int lane = t & 31;
    const int l    = lane & 15;
    const int koff = (lane >> 4) << 1;        // half-wave k offset: 0 or 2
    const int Mbase = (wave >> 2) << 4;       // batch tile base
    const int Nbase = (wave & 3) << 4;        // c_out tile base

    const int m  = t & 63;                    // batch index for A staging
    const int cl = t >> 6;                    // local channel 0..7 for A staging

    const float* Ws = W + (size_t)s * K_TOT * C_OUT;

    // wave-0 test the compiler provably sees as uniform -> scalar branch, so
    // the TDM instruction (which ignores EXEC) is issued by exactly one wave.
    const bool leader = (__builtin_amdgcn_readfirstlane(t) < 32);

    const unsigned int wb_off[2] = {
        (unsigned int)(uintptr_t)&Wbuf[0][0],   // low 32 bits of LDS aperture
        (unsigned int)(uintptr_t)&Wbuf[1][0]    //  = byte offset in LDS
    };

    // ---- prologue: prime both weight buffers via TDM ----
    if (leader) {
        tdm_load_w_chunk(Ws,                 wb_off[0]);   // chunk 0
        tdm_load_w_chunk(Ws + KC * C_OUT,    wb_off[1]);   // chunk 1
    }

    v8f acc = {};

    for (int ci = 0; ci < NCHUNK; ++ci) {
        // ---- stage A (unfold + bias): one (channel,batch) per thread, 9 taps ----
        {
            const int c = ci * NCH + cl;
            const float* Xc = X + ((size_t)m * C_IN + c) * 1024;
            #pragma unroll
            for (int i = 0; i < 3; ++i) {
                const int h  = y + i - 1;
                const bool hok = ((unsigned)h < 32u);
                #pragma unroll
                for (int j = 0; j < 3; ++j) {
                    const int w = x + j - 1;
                    float v = (hok && (unsigned)w < 32u) ? Xc[h * 32 + w] : 0.0f;
                    Alds[m][cl * 9 + i * 3 + j] = v + bias_s;   // bias on all taps
                }
            }
        }
        // ---- chunk ci TDM complete? (in-order per wave: <=1 outstanding left) ----
        if (leader) {
            if (ci == NCHUNK - 1) __builtin_amdgcn_s_wait_tensorcnt(0);
            else                  __builtin_amdgcn_s_wait_tensorcnt(1);
        }
        __syncthreads();   // A staged + weight chunk ci visible to all waves

        // ---- 18 x V_WMMA_F32_16X16X4_F32 over this chunk ----
        const float* Arow = &Alds[Mbase + l][0];
        const float* Wb   = &Wbuf[ci & 1][Nbase + l];
        #pragma unroll
        for (int kk = 0; kk < KC; kk += 4) {
            v2f a = *(const v2f*)(Arow + kk + koff);          // ds_load_b64
            v2f b;
            b.x = Wb[(kk + koff)     * WROW];                 // row k,   bank-shifted
            b.y = Wb[(kk + koff + 1) * WROW];                 // row k+1
            acc = __builtin_amdgcn_wmma_f32_16x16x4_f32(
                    false, a, false, b, (short)0, acc, false, false);
        }
        __syncthreads();   // everyone done with Wbuf[ci&1] and Alds

        // ---- refill the freed buffer with chunk ci+2 ----
        if (leader && (ci + 2) < NCHUNK) {
            tdm_load_w_chunk(Ws + (size_t)(ci + 2) * KC * C_OUT, wb_off[ci & 1]);
        }
    }

    // ---- store per C/D layout: VGPR r -> M = Mbase + r (+8 for upper half) ----
    const int half = lane >> 4;
    #pragma unroll
    for (int r = 0; r < 8; ++r) {
        const int b = Mbase + r + (half << 3);
        const int c = Nbase + l;
        Out[(((size_t)b * C_OUT + c) << 10) + s] = acc[r];
    }
}

extern "C" void kernel_launch(void* const* d_in, const int* in_sizes, int n_in,
                              void* d_out, int out_size, void* d_ws, size_t ws_size,
                              hipStream_t stream) {
    const float* X    = (const float*)d_in[0];   // (64,128,32,32)
    const float* Wgt  = (const float*)d_in[1];   // (1024,1152,64)
    const float* Bias = (const float*)d_in[2];   // (32,32)
    float* Out        = (float*)d_out;           // (64,64,32,32)
    (void)in_sizes; (void)n_in; (void)out_size; (void)d_ws; (void)ws_size;

    varconv_wmma_f32<<<dim3(S_TOT), dim3(512), 0, stream>>>(X, Wgt, Bias, Out);
}